// SIR_NeuralODEModel_22720376996223
// MI455X (gfx1250) — compile-verified
//
#include <hip/hip_runtime.h>
#include <hip/hip_bf16.h>

typedef __attribute__((ext_vector_type(2))) float v2f;
typedef __attribute__((ext_vector_type(8))) float v8f;

#define HIDDEN   128
#define SEQ_LEN  50
#define PRED_LEN 50
#define NSTATE   3
#define BATCH    131072

// One wave (32 lanes) integrates 16 ODE states.
// Transposed formulation:
//   H^T(128x16) = W1^T(128x4) * Y^T(4x16)      [K=3 slot carries b1 via 1.0]
//   O^T(16x16)  = W2^T(16x128) * H^T(128x16)   [rows 0..2 valid, b2 in C init]
// H^T D-tiles are consumed directly as layer-2 B operands (K-permuted),
// with W2^T A operands pre-built in the matching permutation.
__global__ __launch_bounds__(256) void node_rk4_wmma(
    const float* __restrict__ x,
    const float* __restrict__ W1,   // (3,128) row-major
    const float* __restrict__ b1,   // (128)
    const float* __restrict__ W2,   // (128,3) row-major
    const float* __restrict__ b2,   // (3)
    float* __restrict__ out)        // (BATCH, 50, 3)
{
    const int lane   = threadIdx.x & 31;
    const int wave   = (blockIdx.x * blockDim.x + threadIdx.x) >> 5;
    const int b0     = wave * 16;
    const int lo     = lane & 15;
    const bool isLo  = lane < 16;

    // ---- Layer-1 A operands: W1^T tiles (16 hidden x K4), bias folded at K=3 ----
    // f32 A layout: vgpr0 = {K0 lanes0-15, K2 lanes16-31}, vgpr1 = {K1, K3}
    v2f w1a[8];
#pragma unroll
    for (int t = 0; t < 8; ++t) {
        const int h = 16 * t + lo;
        w1a[t][0] = isLo ? W1[0 * HIDDEN + h] : W1[2 * HIDDEN + h];
        w1a[t][1] = isLo ? W1[1 * HIDDEN + h] : b1[h];
    }

    // ---- Layer-2 A operands: W2^T chunks matching H^T D-register K-permutation.
    // Chunk (t,c): K0->h=16t+2c, K1->16t+2c+1, K2->16t+2c+8, K3->16t+2c+9
    v2f w2a[8][4];
#pragma unroll
    for (int t = 0; t < 8; ++t)
#pragma unroll
        for (int c = 0; c < 4; ++c) {
            const int hA = 16 * t + 2 * c + (isLo ? 0 : 8); // K0 / K2
            const int hB = hA + 1;                          // K1 / K3
            const int m  = lo;                              // state row of W2^T
            w2a[t][c][0] = (m < NSTATE) ? W2[hA * NSTATE + m] : 0.0f;
            w2a[t][c][1] = (m < NSTATE) ? W2[hB * NSTATE + m] : 0.0f;
        }

    // ---- Layer-2 accumulator init with b2 (C/D layout: vgpr r, lanes0-15 = row r)
    v8f oinit = (v8f)0.0f;
    if (isLo) { oinit[0] = b2[0]; oinit[1] = b2[1]; oinit[2] = b2[2]; }

    // ---- Load y0 = x[b, SEQ_LEN-1, :] into Y^T B layout ----
    // yb0: lanes0-15 = s0, lanes16-31 = s2 ; yb1: lanes0-15 = s1, lanes16-31 = 1.0
    const float* xrow = x + (size_t)(b0 + lo) * (SEQ_LEN * NSTATE) + (SEQ_LEN - 1) * NSTATE;
    float yb0 = isLo ? xrow[0] : xrow[2];
    float yb1 = 1.0f;
    if (isLo) yb1 = xrow[1];

    float* orow = out + (size_t)(b0 + lo) * (PRED_LEN * NSTATE);
    if (isLo) { orow[0] = yb0; orow[1] = yb1; }
    else      { orow[2] = yb0; }

    // ---- MLP stage eval: input state in Y^T B layout, output k in same layout ----
    // Two independent layer-2 accumulators (even/odd tiles) halve the depth of
    // the serially dependent D->C WMMA chain.
    auto mlp = [&](float yv0, float yv1, float& kb0, float& kb1) {
        v2f yB; yB[0] = yv0; yB[1] = yv1;
        v8f oA = oinit;          // carries b2
        v8f oB = (v8f)0.0f;
#pragma unroll
        for (int t = 0; t < 8; ++t) {
            // H^T tile = W1^T_t * Y^T  (bias via K=3), C = 0
            v8f h = __builtin_amdgcn_wmma_f32_16x16x4_f32(
                false, w1a[t], false, yB, (short)0, (v8f)0.0f, false, false);
#pragma unroll
            for (int i = 0; i < 8; ++i)
                h[i] = fmaxf(h[i], 0.0f); // ReLU (dual-issued max pair)
#pragma unroll
            for (int c = 0; c < 4; ++c) {
                v2f hB; hB[0] = h[2 * c]; hB[1] = h[2 * c + 1]; // B operand, K-permuted
                if (t & 1)
                    oB = __builtin_amdgcn_wmma_f32_16x16x4_f32(
                        false, w2a[t][c], false, hB, (short)0, oB, false, false);
                else
                    oA = __builtin_amdgcn_wmma_f32_16x16x4_f32(
                        false, w2a[t][c], false, hB, (short)0, oA, false, false);
            }
        }
        const float o0 = oA[0] + oB[0];
        const float o1 = oA[1] + oB[1];
        const float o2 = oA[2] + oB[2];
        // O^T rows 0..2 -> Y^T B layout: move s2 (lanes0-15) to lanes16-31
        const int srcByte = lo << 2;
        const float s2 = __builtin_bit_cast(
            float, __builtin_amdgcn_ds_bpermute(srcByte, __builtin_bit_cast(int, o2)));
        kb0 = isLo ? o0 : s2;
        kb1 = isLo ? o1 : 0.0f; // keeps the 1.0 bias lane invariant under y += a*k
    };

    const float dt  = 1.0f / (float)(PRED_LEN - 1);
    const float hdt = 0.5f * dt;
    const float dt6 = dt * (1.0f / 6.0f);

    for (int step = 1; step < PRED_LEN; ++step) {
        float k1b0, k1b1, k2b0, k2b1, k3b0, k3b1, k4b0, k4b1;
        mlp(yb0,               yb1,               k1b0, k1b1);
        mlp(yb0 + hdt * k1b0,  yb1 + hdt * k1b1,  k2b0, k2b1);
        mlp(yb0 + hdt * k2b0,  yb1 + hdt * k2b1,  k3b0, k3b1);
        mlp(yb0 + dt  * k3b0,  yb1 + dt  * k3b1,  k4b0, k4b1);
        yb0 += dt6 * (k1b0 + 2.0f * (k2b0 + k3b0) + k4b0);
        yb1 += dt6 * (k1b1 + 2.0f * (k2b1 + k3b1) + k4b1);
        if (isLo) { orow[3 * step + 0] = yb0; orow[3 * step + 1] = yb1; }
        else      { orow[3 * step + 2] = yb0; }
    }
}

extern "C" void kernel_launch(void* const* d_in, const int* in_sizes, int n_in,
                              void* d_out, int out_size, void* d_ws, size_t ws_size,
                              hipStream_t stream) {
    const float* x  = (const float*)d_in[0];
    const float* W1 = (const float*)d_in[1];
    const float* b1 = (const float*)d_in[2];
    const float* W2 = (const float*)d_in[3];
    const float* b2 = (const float*)d_in[4];
    float* out = (float*)d_out;

    // 16 batch elements per wave, 8 waves (256 threads) per block
    const int waves  = BATCH / 16;           // 8192
    const int blocks = waves / 8;            // 1024
    node_rk4_wmma<<<blocks, 256, 0, stream>>>(x, W1, b1, W2, b2, out);
}